// MLPStackMemory_77103252897908
// MI455X (gfx1250) — compile-verified
//
#include <hip/hip_runtime.h>
#include <hip/hip_bf16.h>

// MLPStackMemory, MI455X (gfx1250, wave32).
// Roofline: 256 MB read + 256 MB write of memory_values dominates -> ~22 us at
// 23.3 TB/s. Compute ~0.5 GFLOP total. Design: NT-hinted float4 streaming for
// the memory slab; V_WMMA_F32_16X16X4_F32 for the two 4096x128 @ 128x128 GEMMs.

#define BN_  4096
#define H_   128
#define M_   128
#define E_   128
#define O_   128

typedef __attribute__((ext_vector_type(2))) float v2f;
typedef __attribute__((ext_vector_type(4))) float v4f;
typedef __attribute__((ext_vector_type(8))) float v8f;

// ---------------------------------------------------------------------------
// GEMM: C[R x 128] = act(A[R x 128] @ W[128 x 128] + bias), act = tanh or id.
// One wave per 16x16 output tile; 8 waves (col tiles) per block; K-loop of 32
// v_wmma_f32_16x16x4_f32.
// A-frag (ISA 7.12.2, 32-bit A 16x4): lane m = lane&15, K pair = (lane>>4)*2.
// B-frag (4x16): assumed mirror of A layout: lane n = lane&15, K pair =
// (lane>>4)*2 (compile-only environment; layout assumption documented).
// C/D (32-bit 16x16): n = lane&15, m = (lane>>4)*8 + vgpr.
// ---------------------------------------------------------------------------
template<bool TANH>
__global__ __launch_bounds__(256) void gemm_wmma_k(
    const float* __restrict__ A, const float* __restrict__ W,
    const float* __restrict__ bias, float* __restrict__ C)
{
  const int lane  = threadIdx.x;        // 0..31 (wave32)
  const int ctile = threadIdx.y;        // 0..7 col tile
  const int row0  = blockIdx.x * 16;
  const int col0  = ctile * 16;
  const int r     = lane & 15;
  const int kh    = lane >> 4;          // 0/1 -> K half

  const float* Arow = A + (size_t)(row0 + r) * 128;
  v8f acc = {};
  #pragma unroll 4
  for (int k = 0; k < 128; k += 4) {
    const int kk = k + kh * 2;
    v2f af, bf;
    af[0] = Arow[kk];
    af[1] = Arow[kk + 1];
    bf[0] = W[(size_t)kk * 128 + col0 + r];
    bf[1] = W[(size_t)(kk + 1) * 128 + col0 + r];
    // 8 args: (neg_a, A, neg_b, B, c_mod, C, reuse_a, reuse_b)
    acc = __builtin_amdgcn_wmma_f32_16x16x4_f32(
        false, af, false, bf, (short)0, acc, false, false);
  }
  const float bn = bias[col0 + r];
  #pragma unroll
  for (int v = 0; v < 8; ++v) {
    const int m = kh * 8 + v;
    float x = acc[v] + bn;
    if (TANH) x = tanhf(x);
    C[(size_t)(row0 + m) * 128 + col0 + r] = x;
  }
}

// ---------------------------------------------------------------------------
// Per-row stack update: one wave32 per row, 4 stack slots per lane.
// push/pop = sigmoid(h . w); suffix sums via shfl inclusive scan.
// Emits coeff (ws), new_rs and new_wm (outputs).
// ---------------------------------------------------------------------------
__global__ __launch_bounds__(256) void stack_k(
    const float* __restrict__ nxt_hidden,
    const float* __restrict__ read_strengths,
    const float* __restrict__ write_mask,
    const float* __restrict__ w_push, const float* __restrict__ b_push,
    const float* __restrict__ w_pop,  const float* __restrict__ b_pop,
    float* __restrict__ coeff_ws,
    float* __restrict__ new_rs, float* __restrict__ new_wm)
{
  const int lane = threadIdx.x & 31;
  const int wid  = threadIdx.x >> 5;
  const int b    = blockIdx.x * 8 + wid;   // BN divisible by 8

  // ---- push / pop gates: 128-elem dot products, 4 elems/lane + wave reduce
  const float* h = nxt_hidden + (size_t)b * H_;
  float dpush = 0.f, dpop = 0.f;
  #pragma unroll
  for (int i = 0; i < 4; ++i) {
    const int idx = lane * 4 + i;
    const float hv = h[idx];
    dpush += hv * w_push[idx];
    dpop  += hv * w_pop[idx];
  }
  #pragma unroll
  for (int off = 16; off > 0; off >>= 1) {
    dpush += __shfl_down(dpush, off, 32);
    dpop  += __shfl_down(dpop,  off, 32);
  }
  dpush = __shfl(dpush, 0, 32);
  dpop  = __shfl(dpop,  0, 32);
  const float push = 1.f / (1.f + __expf(-(dpush + b_push[0])));
  const float pop  = 1.f / (1.f + __expf(-(dpop  + b_pop[0])));

  // ---- load s, w (lane owns m = lane*4 .. lane*4+3)
  float s[4], w[4];
  #pragma unroll
  for (int i = 0; i < 4; ++i) {
    s[i] = read_strengths[(size_t)b * M_ + lane * 4 + i];
    w[i] = write_mask[(size_t)b * M_ + lane * 4 + i];
  }

  // sum_above[i] = total - inclusive_prefix[i] over the 128-elem vector
  auto suffix = [&](const float x[4], float sa[4]) {
    const float p0 = x[0];
    const float p1 = p0 + x[1];
    const float p2 = p1 + x[2];
    const float p3 = p2 + x[3];
    float sc = p3;                        // lane total -> inclusive wave scan
    #pragma unroll
    for (int off = 1; off < 32; off <<= 1) {
      const float t = __shfl_up(sc, off, 32);
      if (lane >= off) sc += t;
    }
    const float total = __shfl(sc, 31, 32);
    const float ex = sc - p3;             // exclusive lane base
    sa[0] = total - (ex + p0);
    sa[1] = total - (ex + p1);
    sa[2] = total - (ex + p2);
    sa[3] = total - (ex + p3);
  };

  float sa[4];
  suffix(s, sa);
  float snew[4];
  #pragma unroll
  for (int i = 0; i < 4; ++i) {
    const float spop = fmaxf(s[i] - fmaxf(pop - sa[i], 0.f), 0.f);
    snew[i] = spop + push * w[i];
    new_rs[(size_t)b * M_ + lane * 4 + i] = snew[i];
    // new_wm = roll(w, 1): scatter w[m] to slot (m+1) mod 128
    new_wm[(size_t)b * M_ + ((lane * 4 + i + 1) & (M_ - 1))] = w[i];
  }

  float sa2[4];
  suffix(snew, sa2);
  #pragma unroll
  for (int i = 0; i < 4; ++i) {
    const float c = fminf(snew[i], fmaxf(1.f - sa2[i], 0.f));
    coeff_ws[(size_t)b * M_ + lane * 4 + i] = c;
  }
}

// ---------------------------------------------------------------------------
// Bandwidth kernel: per row b, stream the 64 KB memory slab once.
//   new_mem[m][e] = mem[m][e] + w[m]*vals[e];  read_vals[e] += coeff[m]*new_mem
// 256 threads = 8 m-partials x 32 float4 column-groups; each wave touches a
// full contiguous 512 B row per iteration (perfectly coalesced). Nontemporal
// load/store (slab is 256 MB each way, > 192 MB L2). LDS tree-reduce partials.
// ---------------------------------------------------------------------------
__global__ __launch_bounds__(256) void stream_k(
    const float* __restrict__ mem, const float* __restrict__ wmask,
    const float* __restrict__ vals, const float* __restrict__ coeff,
    float* __restrict__ new_mem, float* __restrict__ read_vals)
{
  const int b  = blockIdx.x;
  const int t  = threadIdx.x;
  const int cg = t & 31;                 // float4 column group: e = cg*4..+3
  const int mp = t >> 5;                 // 0..7 m-partial (== wave id)

  __shared__ float s_c[M_];
  __shared__ float s_w[M_];
  __shared__ v4f   s_red[8][32];

  if (t < M_) {
    s_c[t] = coeff[(size_t)b * M_ + t];
    s_w[t] = wmask[(size_t)b * M_ + t];
  }
  __syncthreads();

  const v4f* mb = (const v4f*)(mem + (size_t)b * M_ * E_);
  v4f*       nm = (v4f*)(new_mem + (size_t)b * M_ * E_);
  const v4f  vv = ((const v4f*)(vals + (size_t)b * E_))[cg];

  v4f acc = {};
  #pragma unroll 2
  for (int m = mp; m < M_; m += 8) {
    if (m + 8 < M_) __builtin_prefetch(&mb[(m + 8) * 32 + cg], 0, 0);
    v4f x = __builtin_nontemporal_load(&mb[m * 32 + cg]);
    const float wv = s_w[m];
    x += wv * vv;                        // rank-1 one-hot update
    acc += s_c[m] * x;                   // coeff-weighted read
    __builtin_nontemporal_store(x, &nm[m * 32 + cg]);
  }
  s_red[mp][cg] = acc;
  __syncthreads();
  if (t < 32) {
    v4f tot = {};
    #pragma unroll
    for (int i = 0; i < 8; ++i) tot += s_red[i][cg];
    ((v4f*)(read_vals + (size_t)b * E_))[cg] = tot;
  }
}

// ---------------------------------------------------------------------------
extern "C" void kernel_launch(void* const* d_in, const int* in_sizes, int n_in,
                              void* d_out, int out_size, void* d_ws, size_t ws_size,
                              hipStream_t stream) {
  const float* nxt_hidden     = (const float*)d_in[0];
  const float* memory_values  = (const float*)d_in[1];
  const float* read_strengths = (const float*)d_in[2];
  const float* write_mask     = (const float*)d_in[3];
  const float* w_push         = (const float*)d_in[4];
  const float* b_push         = (const float*)d_in[5];
  const float* w_pop          = (const float*)d_in[6];
  const float* b_pop          = (const float*)d_in[7];
  const float* w_val          = (const float*)d_in[8];
  const float* b_val          = (const float*)d_in[9];
  const float* w_out          = (const float*)d_in[10];
  const float* b_out          = (const float*)d_in[11];

  // d_out = concat(out, new_mem, new_rs, new_wm), flat f32
  float* out_p     = (float*)d_out;
  float* new_mem_p = out_p + (size_t)BN_ * O_;
  float* new_rs_p  = new_mem_p + (size_t)BN_ * M_ * E_;
  float* new_wm_p  = new_rs_p + (size_t)BN_ * M_;

  // workspace: vals (2 MB) | coeff (2 MB) | read_vals (2 MB)
  float* ws_vals  = (float*)d_ws;
  float* ws_coeff = ws_vals + (size_t)BN_ * E_;
  float* ws_read  = ws_coeff + (size_t)BN_ * M_;

  dim3 gblock(32, 8);
  // 1) vals = tanh(h @ w_val + b_val)   [WMMA f32]
  gemm_wmma_k<true><<<BN_ / 16, gblock, 0, stream>>>(nxt_hidden, w_val, b_val, ws_vals);
  // 2) gates + stack scan -> coeff, new_rs, new_wm
  stack_k<<<BN_ / 8, 256, 0, stream>>>(nxt_hidden, read_strengths, write_mask,
                                       w_push, b_push, w_pop, b_pop,
                                       ws_coeff, new_rs_p, new_wm_p);
  // 3) stream memory slab: new_mem + read_vals
  stream_k<<<BN_, 256, 0, stream>>>(memory_values, write_mask, ws_vals, ws_coeff,
                                    new_mem_p, ws_read);
  // 4) out = read_vals @ w_out + b_out  [WMMA f32]
  gemm_wmma_k<false><<<BN_ / 16, gblock, 0, stream>>>(ws_read, w_out, b_out, out_p);
}